// MolecularConvolutionLayer_14705968022035
// MI455X (gfx1250) — compile-verified
//
#include <hip/hip_runtime.h>

typedef __attribute__((ext_vector_type(2))) float v2f;
typedef __attribute__((ext_vector_type(8))) float v8f;

#define DA   75
#define DP   14
#define DAO  50
#define DPO  50
#define DAGG 32

// K-chunk (K/4) and N-tile (N/16) counts per GEMM (K,N zero-padded in packing)
#define CH_PA 23
#define NT_PA 2
#define CH_AP 23
#define NT_AP 4
#define CH_PP 4
#define NT_PP 4
#define CH_AO 27
#define NT_AO 4
#define CH_AA 19
#define NT_AA 4

#define XWP 92    // staged X width, pair kernel (89 -> 92)
#define XWA 108   // staged X width, atom kernel (107 -> 108)

__device__ __forceinline__ v8f wmma_f32_16x16x4(v2f a, v2f b, v8f c) {
  // 8 args: (neg_a, A, neg_b, B, c_mod, C, reuse_a, reuse_b)
  return __builtin_amdgcn_wmma_f32_16x16x4_f32(false, a, false, b, (short)0, c,
                                               false, false);
}

__global__ __launch_bounds__(256) void zero_f32(float* __restrict__ p, long n) {
  long i = (long)blockIdx.x * 256 + threadIdx.x;
  if (i < n) p[i] = 0.0f;
}

// Pack row-major W[K][N] into WMMA B-fragment order:
// dst[((c*ntiles+t)*32 + lane)*2 + v] = W[k][n], k = c*4 + (lane>=16)*2 + v,
// n = t*16 + (lane&15); out-of-range k/n -> 0 (lets us over-pad K safely).
__global__ __launch_bounds__(256) void pack_weights(const float* __restrict__ src,
                                                    float* __restrict__ dst,
                                                    int K, int N, int chunks,
                                                    int ntiles) {
  int i = blockIdx.x * 256 + threadIdx.x;
  int total = chunks * ntiles * 64;
  if (i >= total) return;
  int v    = i & 1;
  int lane = (i >> 1) & 31;
  int base = i >> 6;
  int t    = base % ntiles;
  int c    = base / ntiles;
  int k = c * 4 + ((lane >> 4) << 1) + v;
  int n = t * 16 + (lane & 15);
  dst[i] = (k < K && n < N) ? src[k * N + n] : 0.0f;
}

__global__ __launch_bounds__(256, 2) void pair_kernel(
    const float* __restrict__ atom_f, const float* __restrict__ pair_f,
    const int* __restrict__ a2p,
    const float* __restrict__ pw_pa, const float* __restrict__ b_pa,
    const float* __restrict__ pw_ap, const float* __restrict__ b_ap,
    const float* __restrict__ pw_pp, const float* __restrict__ b_pp,
    float* __restrict__ pair_out, float* __restrict__ agg, int n_pairs) {
  __shared__ float xs[8][16 * XWP];
  __shared__ int si[8][16];
  const int wave  = threadIdx.x >> 5;
  const int lane  = threadIdx.x & 31;
  const long tile = (long)blockIdx.x * 8 + wave;
  const long p0   = tile * 16;
  const int mbase = (lane >> 4) << 3;  // 0 or 8: C rows owned by this lane
  const int ncol  = lane & 15;         // C column (within n-tile)
  const int arow  = lane & 15;         // A row owned by this lane
  const int koff  = (lane >> 4) << 1;  // A k-offset within chunk: 0 or 2

  if (lane < 16) {
    long p = p0 + lane;
    si[wave][lane] = (p < n_pairs) ? a2p[2 * p] : -1;
  }

  // ---- Phase 1: X = [pair_f | atom_i + atom_j], padded to XWP ----
  for (int it = 0; it < (16 * XWP) / 32; ++it) {
    int idx = it * 32 + lane;
    int row = idx / XWP;
    int col = idx - row * XWP;
    long p  = p0 + row;
    long pc = (p < n_pairs) ? p : (long)(n_pairs - 1);
    float v = 0.0f;
    if (col < DP) {
      v = pair_f[pc * DP + col];
    } else if (col < DP + DA) {
      int ai = a2p[2 * pc];
      int aj = a2p[2 * pc + 1];
      int c  = col - DP;
      v = atom_f[(long)ai * DA + c] + atom_f[(long)aj * DA + c];
    }
    xs[wave][idx] = v;
  }
  __syncthreads();

  const float* xrow = &xs[wave][0];

  v8f acc_ap[NT_AP] = {};
  v8f acc_pp[NT_PP] = {};
  for (int c = 0; c < CH_AP; ++c) {
    v2f a = *(const v2f*)&xrow[arow * XWP + c * 4 + koff];
#pragma unroll
    for (int t = 0; t < NT_AP; ++t) {
      v2f b = *(const v2f*)&pw_ap[((c * NT_AP + t) * 32 + lane) * 2];
      acc_ap[t] = wmma_f32_16x16x4(a, b, acc_ap[t]);
    }
  }
  // W_pp packed with K padded to 16, rows 14..15 zeroed -> safe to read X[0..15]
  for (int c = 0; c < CH_PP; ++c) {
    v2f a = *(const v2f*)&xrow[arow * XWP + c * 4 + koff];
#pragma unroll
    for (int t = 0; t < NT_PP; ++t) {
      v2f b = *(const v2f*)&pw_pp[((c * NT_PP + t) * 32 + lane) * 2];
      acc_pp[t] = wmma_f32_16x16x4(a, b, acc_pp[t]);
    }
  }

  // pair_hidden = relu(relu(X@W_ap+b) + relu(pair_f@W_pp+b)); 320MB -> NT store
#pragma unroll
  for (int t = 0; t < NT_AP; ++t) {
    int n = t * 16 + ncol;
    float ba = (n < DPO) ? b_ap[n] : 0.0f;
    float bp = (n < DPO) ? b_pp[n] : 0.0f;
#pragma unroll
    for (int g = 0; g < 8; ++g) {
      int m  = mbase + g;
      long p = p0 + m;
      float v = fmaxf(acc_ap[t][g] + ba, 0.0f) + fmaxf(acc_pp[t][g] + bp, 0.0f);
      v = fmaxf(v, 0.0f);
      if (p < n_pairs && n < DPO)
        __builtin_nontemporal_store(v, &pair_out[p * DPO + n]);
    }
  }

  // ---- Phase 2: X = [pair_f | atom_j] ----
  __syncthreads();
  for (int it = 0; it < (16 * XWP) / 32; ++it) {
    int idx = it * 32 + lane;
    int row = idx / XWP;
    int col = idx - row * XWP;
    long p  = p0 + row;
    long pc = (p < n_pairs) ? p : (long)(n_pairs - 1);
    float v = 0.0f;
    if (col < DP) {
      v = pair_f[pc * DP + col];
    } else if (col < DP + DA) {
      int aj = a2p[2 * pc + 1];
      v = atom_f[(long)aj * DA + (col - DP)];
    }
    xs[wave][idx] = v;
  }
  __syncthreads();

  v8f acc_pa[NT_PA] = {};
  for (int c = 0; c < CH_PA; ++c) {
    v2f a = *(const v2f*)&xrow[arow * XWP + c * 4 + koff];
#pragma unroll
    for (int t = 0; t < NT_PA; ++t) {
      v2f b = *(const v2f*)&pw_pa[((c * NT_PA + t) * 32 + lane) * 2];
      acc_pa[t] = wmma_f32_16x16x4(a, b, acc_pa[t]);
    }
  }

  // segment_sum over pair_i (sorted): run-length compress within the 8 rows
  // each lane owns, then one f32 atomic per run boundary.
#pragma unroll
  for (int t = 0; t < NT_PA; ++t) {
    int n = t * 16 + ncol;  // n in [0,32)
    float bb = b_pa[n];
    float run = 0.0f;
    int cur = -1;
#pragma unroll
    for (int g = 0; g < 8; ++g) {
      int m  = mbase + g;
      long p = p0 + m;
      if (p < n_pairs) {
        float v = fmaxf(acc_pa[t][g] + bb, 0.0f);
        int s = si[wave][m];
        if (s == cur) {
          run += v;
        } else {
          if (cur >= 0) atomicAdd(&agg[(long)cur * DAGG + n], run);
          cur = s;
          run = v;
        }
      }
    }
    if (cur >= 0) atomicAdd(&agg[(long)cur * DAGG + n], run);
  }
}

__global__ __launch_bounds__(256, 2) void atom_kernel(
    const float* __restrict__ atom_f, const float* __restrict__ agg,
    const float* __restrict__ pw_ao, const float* __restrict__ b_ao,
    const float* __restrict__ pw_aa, const float* __restrict__ b_aa,
    float* __restrict__ atom_out, int n_atoms) {
  __shared__ float xs[8][16 * XWA];
  const int wave  = threadIdx.x >> 5;
  const int lane  = threadIdx.x & 31;
  const long tile = (long)blockIdx.x * 8 + wave;
  const long a0   = tile * 16;
  const int mbase = (lane >> 4) << 3;
  const int ncol  = lane & 15;
  const int arow  = lane & 15;
  const int koff  = (lane >> 4) << 1;

  for (int it = 0; it < (16 * XWA) / 32; ++it) {
    int idx = it * 32 + lane;
    int row = idx / XWA;
    int col = idx - row * XWA;
    long a  = a0 + row;
    long ac = (a < n_atoms) ? a : (long)(n_atoms - 1);
    float v = 0.0f;
    if (col < DA) v = atom_f[ac * DA + col];
    else if (col < DA + DAGG) v = agg[ac * DAGG + (col - DA)];
    xs[wave][idx] = v;
  }
  __syncthreads();

  const float* xrow = &xs[wave][0];
  v8f acc_ao[NT_AO] = {};
  v8f acc_aa[NT_AA] = {};
  for (int c = 0; c < CH_AO; ++c) {
    v2f a = *(const v2f*)&xrow[arow * XWA + c * 4 + koff];
#pragma unroll
    for (int t = 0; t < NT_AO; ++t) {
      v2f b = *(const v2f*)&pw_ao[((c * NT_AO + t) * 32 + lane) * 2];
      acc_ao[t] = wmma_f32_16x16x4(a, b, acc_ao[t]);
    }
  }
  // W_aa packed with K padded to 76, row 75 zeroed -> safe to read X[0..75]
  for (int c = 0; c < CH_AA; ++c) {
    v2f a = *(const v2f*)&xrow[arow * XWA + c * 4 + koff];
#pragma unroll
    for (int t = 0; t < NT_AA; ++t) {
      v2f b = *(const v2f*)&pw_aa[((c * NT_AA + t) * 32 + lane) * 2];
      acc_aa[t] = wmma_f32_16x16x4(a, b, acc_aa[t]);
    }
  }

#pragma unroll
  for (int t = 0; t < NT_AO; ++t) {
    int n = t * 16 + ncol;
    float bo = (n < DAO) ? b_ao[n] : 0.0f;
    float ba = (n < DAO) ? b_aa[n] : 0.0f;
#pragma unroll
    for (int g = 0; g < 8; ++g) {
      int m  = mbase + g;
      long a = a0 + m;
      float v = fmaxf(acc_ao[t][g] + bo, 0.0f) + fmaxf(acc_aa[t][g] + ba, 0.0f);
      v = fmaxf(v, 0.0f);
      if (a < n_atoms && n < DAO) atom_out[a * DAO + n] = v;
    }
  }
}

extern "C" void kernel_launch(void* const* d_in, const int* in_sizes, int n_in,
                              void* d_out, int out_size, void* d_ws,
                              size_t ws_size, hipStream_t stream) {
  const float* atom_f = (const float*)d_in[0];
  const float* pair_f = (const float*)d_in[1];
  const float* W_pa = (const float*)d_in[2];
  const float* b_pa = (const float*)d_in[3];
  const float* W_ao = (const float*)d_in[4];
  const float* b_ao = (const float*)d_in[5];
  const float* W_aa = (const float*)d_in[6];
  const float* b_aa = (const float*)d_in[7];
  const float* W_ap = (const float*)d_in[8];
  const float* b_ap = (const float*)d_in[9];
  const float* W_pp = (const float*)d_in[10];
  const float* b_pp = (const float*)d_in[11];
  const int* a2p = (const int*)d_in[13];
  const int n_atoms = in_sizes[0] / DA;
  const int n_pairs = in_sizes[1] / DP;

  float* ws  = (float*)d_ws;
  float* agg = ws;  // [n_atoms][32] segment sums (atomic)
  size_t off = (size_t)n_atoms * DAGG;
  float* pw_pa = ws + off; off += CH_PA * NT_PA * 64;
  float* pw_ap = ws + off; off += CH_AP * NT_AP * 64;
  float* pw_pp = ws + off; off += CH_PP * NT_PP * 64;
  float* pw_ao = ws + off; off += CH_AO * NT_AO * 64;
  float* pw_aa = ws + off; off += CH_AA * NT_AA * 64;

  float* atom_out = (float*)d_out;
  float* pair_out = atom_out + (size_t)n_atoms * DAO;

  long aggN = (long)n_atoms * DAGG;
  zero_f32<<<(int)((aggN + 255) / 256), 256, 0, stream>>>(agg, aggN);
  pack_weights<<<(CH_PA * NT_PA * 64 + 255) / 256, 256, 0, stream>>>(
      W_pa, pw_pa, DP + DA, DAGG, CH_PA, NT_PA);
  pack_weights<<<(CH_AP * NT_AP * 64 + 255) / 256, 256, 0, stream>>>(
      W_ap, pw_ap, DP + DA, DPO, CH_AP, NT_AP);
  pack_weights<<<(CH_PP * NT_PP * 64 + 255) / 256, 256, 0, stream>>>(
      W_pp, pw_pp, DP, DPO, CH_PP, NT_PP);
  pack_weights<<<(CH_AO * NT_AO * 64 + 255) / 256, 256, 0, stream>>>(
      W_ao, pw_ao, DA + DAGG, DAO, CH_AO, NT_AO);
  pack_weights<<<(CH_AA * NT_AA * 64 + 255) / 256, 256, 0, stream>>>(
      W_aa, pw_aa, DA, DAO, CH_AA, NT_AA);

  int ptiles  = (n_pairs + 15) / 16;
  int pblocks = (ptiles + 7) / 8;
  pair_kernel<<<pblocks, 256, 0, stream>>>(atom_f, pair_f, a2p, pw_pa, b_pa,
                                           pw_ap, b_ap, pw_pp, b_pp, pair_out,
                                           agg, n_pairs);

  int atiles  = (n_atoms + 15) / 16;
  int ablocks = (atiles + 7) / 8;
  atom_kernel<<<ablocks, 256, 0, stream>>>(atom_f, agg, pw_ao, b_ao, pw_aa,
                                           b_aa, atom_out, n_atoms);
}